// NeighborAdjustingLoss_33500744908968
// MI455X (gfx1250) — compile-verified
//
#include <hip/hip_runtime.h>
#include <math.h>

// Problem constants (fixed by the reference's setup_inputs).
#define BDIM   4096
#define MDIM   32768
#define KNB    32
#define CSPL   8                      // column slices for the streaming kernel
#define SLICE  (MDIM / CSPL)          // 4096 columns per slice
#define TW     512                    // tile width (columns) staged in LDS
#define TP     516                    // padded row stride (floats): 516 % 64 == 4 -> conflict-free

#if __has_builtin(__builtin_amdgcn_global_load_async_to_lds_b128) && \
    __has_builtin(__builtin_amdgcn_s_wait_asynccnt)
#define HAS_ASYNC_LDS 1
#else
#define HAS_ASYNC_LDS 0
#endif

typedef float v2f __attribute__((ext_vector_type(2)));
typedef float v8f __attribute__((ext_vector_type(8)));

#if HAS_ASYNC_LDS
typedef int v4i __attribute__((vector_size(16)));
typedef v4i __attribute__((address_space(1))) *global_v4i_ptr;
typedef v4i __attribute__((address_space(3))) *lds_v4i_ptr;
#endif

// ---------------------------------------------------------------------------
// Kernel 1: partial row sums of mb_mat via V_WMMA_F32_16X16X4_F32.
// Each block: 16 rows x SLICE cols. 8 waves; wave w owns 64 cols of each tile.
// B matrix = all ones  =>  D[m][n] = sum_k A[m][k] + C[m][n]  (layout-proof).
// A layout (32-bit 16x4): lanes 0-15 hold M=lane, K=0/1; lanes 16-31 M=lane-16, K=2/3.
// Data movement: double-buffered GLOBAL_LOAD_ASYNC_TO_LDS_B128 (ASYNCcnt),
// overlapped with the WMMA pipe; falls back to sync b128 copies if the
// toolchain lacks the async builtin.
// ---------------------------------------------------------------------------
__global__ __launch_bounds__(256) void centrality_partial_kernel(
    const float* __restrict__ mb, float* __restrict__ part)
{
#if HAS_ASYNC_LDS
    __shared__ float tile[2][16][TP];
#else
    __shared__ float tile[1][16][TP];
#endif
    __shared__ float warp_part[8][16];

    const int wave    = threadIdx.x >> 5;
    const int lane    = threadIdx.x & 31;
    const int rowBase = blockIdx.x * 16;
    const int colBase = blockIdx.y * SLICE;

    const float* mbBase = mb + (size_t)rowBase * MDIM + colBase;

    v8f acc = {0.f, 0.f, 0.f, 0.f, 0.f, 0.f, 0.f, 0.f};
    v2f ones; ones.x = 1.0f; ones.y = 1.0f;

    const int r2   = lane & 15;          // A-matrix row this lane feeds
    const int koff = (lane >> 4) << 1;   // lanes 16-31 supply K=2,3

#if HAS_ASYNC_LDS
    // Issue one 16x512 tile as 8 per-thread async B128 DMAs (coalesced:
    // consecutive lanes hit consecutive 16B chunks of the same row).
    auto issue_tile = [&](int buf, int c0) {
#pragma unroll
        for (int i = 0; i < 8; ++i) {
            int f  = threadIdx.x + (i << 8);  // 0..2047 float4 chunks
            int r  = f >> 7;                  // 128 float4 per row
            int c4 = f & 127;
            const float* g = mbBase + (size_t)r * MDIM + c0 + (c4 << 2);
            float*       l = &tile[buf][r][c4 << 2];
            __builtin_amdgcn_global_load_async_to_lds_b128(
                (global_v4i_ptr)(size_t)g,
                (lds_v4i_ptr)(uint32_t)(size_t)l,
                0, 0);
        }
    };

    constexpr int NT = SLICE / TW;  // 8 tiles per block
    issue_tile(0, 0);
    __builtin_amdgcn_s_wait_asynccnt(0);
    __syncthreads();

    for (int t = 0; t < NT; ++t) {
        const int cur = t & 1;
        if (t + 1 < NT) issue_tile((t + 1) & 1, (t + 1) * TW);

        const int cwave = wave << 6;  // 64 columns per wave
#pragma unroll
        for (int kk = 0; kk < 16; ++kk) {
            const int k0 = cwave + (kk << 2);
            v2f a = *reinterpret_cast<const v2f*>(&tile[cur][r2][k0 + koff]);
            acc = __builtin_amdgcn_wmma_f32_16x16x4_f32(
                false, a, false, ones, (short)0, acc, false, false);
        }
        __builtin_amdgcn_s_wait_asynccnt(0);
        __syncthreads();  // next-tile writes landed; current-tile reads done
    }
#else
    for (int c0 = 0; c0 < SLICE; c0 += TW) {
        // Cooperative coalesced load: 16 x 512 floats = 2048 float4, 8 per thread.
#pragma unroll
        for (int i = 0; i < 8; ++i) {
            int f  = threadIdx.x + (i << 8);
            int r  = f >> 7;
            int c4 = f & 127;
            const float4 v =
                reinterpret_cast<const float4*>(mbBase + (size_t)r * MDIM + c0)[c4];
            *reinterpret_cast<float4*>(&tile[0][r][c4 << 2]) = v;
        }
        __syncthreads();

        const int cwave = wave << 6;
#pragma unroll
        for (int kk = 0; kk < 16; ++kk) {
            const int k0 = cwave + (kk << 2);
            v2f a = *reinterpret_cast<const v2f*>(&tile[0][r2][k0 + koff]);
            acc = __builtin_amdgcn_wmma_f32_16x16x4_f32(
                false, a, false, ones, (short)0, acc, false, false);
        }
        __syncthreads();
    }
#endif

    // D layout: VGPR r = M=r (lanes 0-15, col N=lane) / M=8+r (lanes 16-31).
    // All columns are identical (B = ones), so lane 0 / lane 16 suffice.
    if (lane == 0) {
#pragma unroll
        for (int r = 0; r < 8; ++r) warp_part[wave][r] = acc[r];
    } else if (lane == 16) {
#pragma unroll
        for (int r = 0; r < 8; ++r) warp_part[wave][8 + r] = acc[r];
    }
    __syncthreads();

    if (threadIdx.x < 16) {
        float s = 0.f;
#pragma unroll
        for (int q = 0; q < 8; ++q) s += warp_part[q][threadIdx.x];
        part[(size_t)blockIdx.y * BDIM + rowBase + threadIdx.x] = s;
    }
}

// Kernel 1b: combine the 8 column-slice partials into c[j] = rowmean(mb).
__global__ __launch_bounds__(256) void centrality_combine_kernel(
    const float* __restrict__ part, float* __restrict__ c)
{
    const int j = blockIdx.x * 256 + threadIdx.x;
    float s = 0.f;
#pragma unroll
    for (int q = 0; q < CSPL; ++q) s += part[(size_t)q * BDIM + j];
    c[j] = s * (1.0f / (float)MDIM);
}

// ---------------------------------------------------------------------------
// Kernel 2: one block per row. Top-32, masked min/max, softmax, masked
// log-softmax contribution.
// ---------------------------------------------------------------------------
__global__ __launch_bounds__(256) void row_topk_loss_kernel(
    const float* __restrict__ sim, const float* __restrict__ cvec,
    const float* __restrict__ temp, float* __restrict__ losses)
{
    __shared__ float w[BDIM];      // working copy of sim row; -INF marks excluded
    __shared__ float cv[BDIM];     // centrality vector
    __shared__ float topVal[KNB];
    __shared__ int   topIdx[KNB];
    __shared__ float redV[8];
    __shared__ int   redI[8];
    __shared__ float redMnS[8], redMxS[8], redMnC[8], redMxC[8];
    __shared__ float sh_diag;

    const int tid  = threadIdx.x;
    const int wave = tid >> 5;
    const int lane = tid & 31;
    const int row  = blockIdx.x;
    const float NEG_INF = -INFINITY;

    // Coalesced float4 load of sim row + centrality vector into LDS.
    const float4* simRow4 = reinterpret_cast<const float4*>(sim + (size_t)row * BDIM);
    const float4* cvec4   = reinterpret_cast<const float4*>(cvec);
    float4* w4  = reinterpret_cast<float4*>(w);
    float4* cv4 = reinterpret_cast<float4*>(cv);
#pragma unroll
    for (int i = 0; i < (BDIM / 4) / 256; ++i) {  // 4 iterations
        int f = tid + (i << 8);
        w4[f]  = simRow4[f];
        cv4[f] = cvec4[f];
    }
    __syncthreads();
    if (tid == 0) { sh_diag = w[row]; w[row] = NEG_INF; }
    __syncthreads();

    // Iterative top-K argmax (value desc, ties -> lower index, like top_k).
    for (int it = 0; it < KNB; ++it) {
        float bv = NEG_INF;
        int   bi = 0x7fffffff;
#pragma unroll
        for (int i = 0; i < BDIM / 256; ++i) {
            int j   = tid + (i << 8);      // bank = tid%64: conflict-free
            float v = w[j];
            if (v > bv || (v == bv && j < bi)) { bv = v; bi = j; }
        }
#pragma unroll
        for (int off = 16; off > 0; off >>= 1) {
            float ov = __shfl_xor(bv, off, 32);
            int   oi = __shfl_xor(bi, off, 32);
            if (ov > bv || (ov == bv && oi < bi)) { bv = ov; bi = oi; }
        }
        if (lane == 0) { redV[wave] = bv; redI[wave] = bi; }
        __syncthreads();
        if (tid == 0) {
            float fv = redV[0]; int fi = redI[0];
            for (int q = 1; q < 8; ++q) {
                float qv = redV[q]; int qi = redI[q];
                if (qv > fv || (qv == fv && qi < fi)) { fv = qv; fi = qi; }
            }
            topVal[it] = fv; topIdx[it] = fi;
            w[fi] = NEG_INF;  // remove from further consideration
        }
        __syncthreads();
    }

    // Masked min/max over ext_mask==0 entries (w[j] != -INF) for sim and centrality.
    float mnS = INFINITY, mxS = -INFINITY, mnC = INFINITY, mxC = -INFINITY;
#pragma unroll
    for (int i = 0; i < BDIM / 256; ++i) {
        int j   = tid + (i << 8);
        float v = w[j];
        if (v != NEG_INF) {
            mnS = fminf(mnS, v);
            mxS = fmaxf(mxS, v);
            float cc = cv[j];
            mnC = fminf(mnC, cc);
            mxC = fmaxf(mxC, cc);
        }
    }
#pragma unroll
    for (int off = 16; off > 0; off >>= 1) {
        mnS = fminf(mnS, __shfl_xor(mnS, off, 32));
        mxS = fmaxf(mxS, __shfl_xor(mxS, off, 32));
        mnC = fminf(mnC, __shfl_xor(mnC, off, 32));
        mxC = fmaxf(mxC, __shfl_xor(mxC, off, 32));
    }
    if (lane == 0) {
        redMnS[wave] = mnS; redMxS[wave] = mxS;
        redMnC[wave] = mnC; redMxC[wave] = mxC;
    }
    __syncthreads();

    if (tid == 0) {
        float sMin = redMnS[0], sMax = redMxS[0], cMin = redMnC[0], cMax = redMxC[0];
        for (int q = 1; q < 8; ++q) {
            sMin = fminf(sMin, redMnS[q]); sMax = fmaxf(sMax, redMxS[q]);
            cMin = fminf(cMin, redMnC[q]); cMax = fmaxf(cMax, redMxC[q]);
        }
        const float diag = sh_diag;

        // masked log-softmax over {diag} ∪ top-K
        float m  = fmaxf(diag, topVal[0]);
        float es = expf(diag - m);
        for (int t = 0; t < KNB; ++t) es += expf(topVal[t] - m);
        const float lse = m + logf(es);

        const float T  = temp[0];
        const float dS = sMax - sMin + 1e-8f;
        const float dC = cMax - cMin + 1e-8f;

        float adj[KNB];
        float amax = -INFINITY;
        for (int t = 0; t < KNB; ++t) {
            float ns = (topVal[t] - sMin) / dS;
            float nc = (cv[topIdx[t]] - cMin) / dC;
            float a  = T * (ns - nc);
            adj[t] = a;
            amax   = fmaxf(amax, a);
        }
        float S = 0.f;
        for (int t = 0; t < KNB; ++t) { float e = expf(adj[t] - amax); adj[t] = e; S += e; }

        float contrib = diag - lse;  // pos_w diag == 1
        float denom   = 1.0f;
        for (int t = 0; t < KNB; ++t) {
            float wt = adj[t] / S;
            contrib += wt * (topVal[t] - lse);
            denom   += wt;
        }
        losses[row] = -(contrib / denom);
    }
}

// ---------------------------------------------------------------------------
// Kernel 3: mean of per-row losses -> scalar output.
// ---------------------------------------------------------------------------
__global__ __launch_bounds__(256) void final_mean_kernel(
    const float* __restrict__ losses, float* __restrict__ out)
{
    __shared__ float red[8];
    const int wave = threadIdx.x >> 5;
    const int lane = threadIdx.x & 31;

    float s = 0.f;
#pragma unroll
    for (int i = 0; i < BDIM / 256; ++i) s += losses[threadIdx.x + (i << 8)];
#pragma unroll
    for (int off = 16; off > 0; off >>= 1) s += __shfl_xor(s, off, 32);
    if (lane == 0) red[wave] = s;
    __syncthreads();
    if (threadIdx.x == 0) {
        float t = 0.f;
#pragma unroll
        for (int q = 0; q < 8; ++q) t += red[q];
        out[0] = t * (1.0f / (float)BDIM);
    }
}

// ---------------------------------------------------------------------------
extern "C" void kernel_launch(void* const* d_in, const int* in_sizes, int n_in,
                              void* d_out, int out_size, void* d_ws, size_t ws_size,
                              hipStream_t stream) {
    const float* sim  = (const float*)d_in[0];   // (4096, 4096) f32
    const float* mb   = (const float*)d_in[1];   // (4096, 32768) f32
    const float* temp = (const float*)d_in[2];   // scalar f32
    (void)in_sizes; (void)n_in; (void)out_size; (void)ws_size;

    float* ws     = (float*)d_ws;
    float* c      = ws;                 // [BDIM]
    float* losses = ws + BDIM;          // [BDIM]
    float* part   = ws + 2 * BDIM;      // [CSPL][BDIM]

    centrality_partial_kernel<<<dim3(BDIM / 16, CSPL), 256, 0, stream>>>(mb, part);
    centrality_combine_kernel<<<BDIM / 256, 256, 0, stream>>>(part, c);
    row_topk_loss_kernel<<<BDIM, 256, 0, stream>>>(sim, c, temp, losses);
    final_mean_kernel<<<1, 256, 0, stream>>>(losses, (float*)d_out);
}